// MemoryModule_55722905699030
// MI455X (gfx1250) — compile-verified
//
#include <hip/hip_runtime.h>
#include <stdint.h>

// TGN GRU memory update for MI455X (gfx1250, wave32).
// f32 WMMA (V_WMMA_F32_16X16X4_F32) + async global->LDS staging (ASYNCcnt)
// + weights repacked fragment-major into d_ws for fully coalesced B loads.

typedef float v2f __attribute__((ext_vector_type(2)));
typedef float v8f __attribute__((ext_vector_type(8)));

#define MEMD 128
#define MSGD 256
#define MSG_LDW 260   // 16x260 f32 per wave, row stride 1040B (16B mult, %64w==4)
#define H_LDW   132   // 16x132 f32 per wave, row stride  528B (16B mult, %64w==4)
#define WAVE_LDS (16 * MSG_LDW + 16 * H_LDW)   // 6272 floats = 25088 B
#define WAVES_PER_BLOCK 4

// Packed weight geometry: one B-fragment (16N x 4K) = 32 lanes * float2 = 256B.
#define NIH_PAIRS (24 * 64 * 32)   // W_ih: 24 N-tiles * 64 K-steps
#define NHH_PAIRS (24 * 32 * 32)   // W_hh: 24 N-tiles * 32 K-steps
#define PACK_FLOATS ((NIH_PAIRS + NHH_PAIRS) * 2)   // 147456 floats = 576 KB

__device__ __forceinline__ float sigm_fast(float x) {
    return 1.0f / (1.0f + __expf(-x));
}
__device__ __forceinline__ float tanh_fast(float x) {
    return 2.0f / (1.0f + __expf(-2.0f * x)) - 1.0f;
}

// gfx1250 async copy: global memory -> LDS, 16B per lane, tracked by ASYNCcnt.
__device__ __forceinline__ void async_g2l_b128(const float* lds_ptr, const float* gptr) {
    uint32_t lds_off = (uint32_t)(uintptr_t)lds_ptr;   // low 32 bits = LDS byte offset
    asm volatile("global_load_async_to_lds_b128 %0, %1, off"
                 :: "v"(lds_off), "v"((uint64_t)(uintptr_t)gptr)
                 : "memory");
}
__device__ __forceinline__ void wait_async0() {
    asm volatile("s_wait_asynccnt 0x0" ::: "memory");
}

// One-time repack of W_ih/W_hh into fragment-major order in d_ws.
__global__ void repack_weights(const float* __restrict__ W_ih,
                               const float* __restrict__ W_hh,
                               float* __restrict__ P) {
    const int p = blockIdx.x * blockDim.x + threadIdx.x;
    if (p < NIH_PAIRS) {
        const int lane = p & 31, frag = p >> 5;
        const int kb = frag & 63, ntile = frag >> 6;
        const int n = ntile * 16 + (lane & 15);
        const int k = kb * 4 + ((lane >> 4) << 1);
        *(v2f*)(P + (size_t)p * 2) = *(const v2f*)(W_ih + (size_t)n * MSGD + k);
    } else if (p < NIH_PAIRS + NHH_PAIRS) {
        const int q = p - NIH_PAIRS;
        const int lane = q & 31, frag = q >> 5;
        const int kb = frag & 31, ntile = frag >> 5;
        const int n = ntile * 16 + (lane & 15);
        const int k = kb * 4 + ((lane >> 4) << 1);
        *(v2f*)(P + (size_t)p * 2) = *(const v2f*)(W_hh + (size_t)n * MEMD + k);
    }
}

template <bool PACKED>
__global__ __launch_bounds__(32 * WAVES_PER_BLOCK, 1)
void tgn_gru_wmma_kernel(const float* __restrict__ memory,
                         const int*   __restrict__ node_ids,
                         const float* __restrict__ messages,
                         const float* __restrict__ W_ih,   // [384, 256] row-major
                         const float* __restrict__ W_hh,   // [384, 128] row-major
                         const float* __restrict__ Wp,     // packed weights (or null)
                         const float* __restrict__ b_ih,   // [384]
                         const float* __restrict__ b_hh,   // [384]
                         float* __restrict__ out,          // [500000, 128]
                         int batch)
{
    __shared__ float lds[WAVES_PER_BLOCK * WAVE_LDS];   // ~100 KB / block

    const int lane = threadIdx.x & 31;
    const int wave = threadIdx.x >> 5;
    const int half = lane >> 4;      // K-pair selector for f32 A/B fragments
    const int col  = lane & 15;      // fragment row (A) / column (B, C/D)

    const int m0 = (blockIdx.x * WAVES_PER_BLOCK + wave) * 16;
    if (m0 + 16 > batch) return;     // wave-uniform guard

    float* msgs = &lds[wave * WAVE_LDS];
    float* hs   = msgs + 16 * MSG_LDW;

    const int myid = node_ids[m0 + col];   // node id of fragment row (lane&15)

    // ---- Stage message tile (16x256) and gathered h tile (16x128) into LDS ----
    #pragma unroll
    for (int r = 0; r < 16; ++r) {
        const float* g = messages + (size_t)(m0 + r) * MSGD + lane * 4;
        float* l = msgs + r * MSG_LDW + lane * 4;
        async_g2l_b128(l, g);                       // cols [0,128)
        async_g2l_b128(l + 128, g + 128);           // cols [128,256)
    }
    #pragma unroll
    for (int r = 0; r < 16; ++r) {
        const int idr = __shfl(myid, r);
        async_g2l_b128(hs + r * H_LDW + lane * 4,
                       memory + (size_t)idr * MEMD + lane * 4);
    }
    wait_async0();

    // Per-lane fragment base pointers.
    const float* msg_frag = msgs + col * MSG_LDW + (half << 1);
    const float* h_frag   = hs   + col * H_LDW   + (half << 1);
    const float* pih = PACKED ? (Wp + (size_t)lane * 2) : nullptr;
    const float* phh = PACKED ? (Wp + (size_t)NIH_PAIRS * 2 + lane * 2) : nullptr;

    // r (tiles 0..7) and z (tiles 8..15) kept in fragment layout, in registers.
    v8f rz[16];

    // ================= Phase A: n = 0..255  -> r, z ====================
    #pragma unroll
    for (int g = 0; g < 4; ++g) {
        const int n0 = g * 64;
        const float* wih = W_ih + (size_t)(n0 + col) * MSGD + (half << 1);
        const float* whh = W_hh + (size_t)(n0 + col) * MEMD + (half << 1);
        v8f acc[4] = {v8f{}, v8f{}, v8f{}, v8f{}};

        #pragma unroll 4
        for (int kb = 0; kb < 64; ++kb) {      // gi: K = 256
            v2f a = *(const v2f*)(msg_frag + kb * 4);
            #pragma unroll
            for (int t = 0; t < 4; ++t) {
                v2f b;
                if constexpr (PACKED)
                    b = *(const v2f*)(pih + (size_t)(((g * 4 + t) * 64 + kb)) * 64);
                else
                    b = *(const v2f*)(wih + (size_t)t * 16 * MSGD + kb * 4);
                acc[t] = __builtin_amdgcn_wmma_f32_16x16x4_f32(
                    false, a, false, b, (short)0, acc[t], false, false);
            }
        }
        #pragma unroll 4
        for (int kb = 0; kb < 32; ++kb) {      // gh: K = 128
            v2f a = *(const v2f*)(h_frag + kb * 4);
            #pragma unroll
            for (int t = 0; t < 4; ++t) {
                v2f b;
                if constexpr (PACKED)
                    b = *(const v2f*)(phh + (size_t)(((g * 4 + t) * 32 + kb)) * 64);
                else
                    b = *(const v2f*)(whh + (size_t)t * 16 * MEMD + kb * 4);
                acc[t] = __builtin_amdgcn_wmma_f32_16x16x4_f32(
                    false, a, false, b, (short)0, acc[t], false, false);
            }
        }
        #pragma unroll
        for (int t = 0; t < 4; ++t) {          // bias + sigmoid, stash fragment
            const int n = n0 + t * 16 + col;
            const float bias = b_ih[n] + b_hh[n];
            v8f s;
            #pragma unroll
            for (int i = 0; i < 8; ++i) s[i] = sigm_fast(acc[t][i] + bias);
            rz[g * 4 + t] = s;
        }
    }

    // ============ Phase B: n = 256..383 -> n-gate, blend, scatter ============
    #pragma unroll
    for (int g = 0; g < 4; ++g) {
        const int n0 = 256 + g * 32;
        const float* wih = W_ih + (size_t)(n0 + col) * MSGD + (half << 1);
        const float* whh = W_hh + (size_t)(n0 + col) * MEMD + (half << 1);
        v8f ai[2] = {v8f{}, v8f{}};   // i_n accumulators (input path)
        v8f ah[2] = {v8f{}, v8f{}};   // h_n accumulators (hidden path)

        #pragma unroll 4
        for (int kb = 0; kb < 64; ++kb) {
            v2f a = *(const v2f*)(msg_frag + kb * 4);
            #pragma unroll
            for (int t = 0; t < 2; ++t) {
                v2f b;
                if constexpr (PACKED)
                    b = *(const v2f*)(pih + (size_t)(((16 + g * 2 + t) * 64 + kb)) * 64);
                else
                    b = *(const v2f*)(wih + (size_t)t * 16 * MSGD + kb * 4);
                ai[t] = __builtin_amdgcn_wmma_f32_16x16x4_f32(
                    false, a, false, b, (short)0, ai[t], false, false);
            }
        }
        #pragma unroll 4
        for (int kb = 0; kb < 32; ++kb) {
            v2f a = *(const v2f*)(h_frag + kb * 4);
            #pragma unroll
            for (int t = 0; t < 2; ++t) {
                v2f b;
                if constexpr (PACKED)
                    b = *(const v2f*)(phh + (size_t)(((16 + g * 2 + t) * 32 + kb)) * 64);
                else
                    b = *(const v2f*)(whh + (size_t)t * 16 * MEMD + kb * 4);
                ah[t] = __builtin_amdgcn_wmma_f32_16x16x4_f32(
                    false, a, false, b, (short)0, ah[t], false, false);
            }
        }

        #pragma unroll
        for (int t = 0; t < 2; ++t) {
            const int tt = g * 2 + t;            // n-gate tile index 0..7
            const int n  = n0 + t * 16 + col;    // global gate index 256..383
            const int j  = tt * 16 + col;        // hidden-dim column 0..127
            const float bi = b_ih[n];
            const float bh = b_hh[n];
            const v8f rfrag = rz[tt];
            const v8f zfrag = rz[8 + tt];
            #pragma unroll
            for (int i = 0; i < 8; ++i) {
                const int row = i + 8 * half;            // fragment row 0..15
                const int id  = __shfl(myid, row);        // node id of that row
                const float hv = hs[row * H_LDW + j];     // h from LDS stage
                const float hn = ah[t][i] + bh;
                const float ivn = ai[t][i] + bi;
                const float ng = tanh_fast(ivn + rfrag[i] * hn);
                const float nh = (1.0f - zfrag[i]) * ng + zfrag[i] * hv;
                out[(size_t)id * MEMD + j] = nh;
            }
        }
    }
}

extern "C" void kernel_launch(void* const* d_in, const int* in_sizes, int n_in,
                              void* d_out, int out_size, void* d_ws, size_t ws_size,
                              hipStream_t stream) {
    (void)n_in; (void)out_size;
    const float* memory   = (const float*)d_in[0];
    const int*   node_ids = (const int*)  d_in[1];
    const float* messages = (const float*)d_in[2];
    const float* W_ih     = (const float*)d_in[3];
    const float* W_hh     = (const float*)d_in[4];
    const float* b_ih     = (const float*)d_in[5];
    const float* b_hh     = (const float*)d_in[6];
    float* out = (float*)d_out;

    // Bulk copy memory -> out (updated rows overwritten by the GRU kernel).
    hipMemcpyAsync(out, memory, (size_t)in_sizes[0] * sizeof(float),
                   hipMemcpyDeviceToDevice, stream);

    const int batch  = in_sizes[1];                         // 131072
    const int rowsPerBlock = 16 * WAVES_PER_BLOCK;          // 64
    const int blocks = (batch + rowsPerBlock - 1) / rowsPerBlock;

    const bool usePacked = (ws_size >= (size_t)PACK_FLOATS * sizeof(float));
    if (usePacked) {
        float* Wp = (float*)d_ws;
        const int pairs = NIH_PAIRS + NHH_PAIRS;
        repack_weights<<<(pairs + 255) / 256, 256, 0, stream>>>(W_ih, W_hh, Wp);
        tgn_gru_wmma_kernel<true><<<blocks, 32 * WAVES_PER_BLOCK, 0, stream>>>(
            memory, node_ids, messages, W_ih, W_hh, Wp, b_ih, b_hh, out, batch);
    } else {
        tgn_gru_wmma_kernel<false><<<blocks, 32 * WAVES_PER_BLOCK, 0, stream>>>(
            memory, node_ids, messages, W_ih, W_hh, nullptr, b_ih, b_hh, out, batch);
    }
}